// QuantumKernel_65481071407639
// MI455X (gfx1250) — compile-verified
//
#include <hip/hip_runtime.h>

// Mathematically, the reference circuit reduces to out[b,w] = cos(x[b,w]):
//  - RZ layers are diagonal phases (unit modulus) -> cannot change |amp|^2
//  - each CNOT chain is the GF(2) prefix-XOR map M on basis bits; M^4 == I
//    for 4 bits, so the 4-layer permutation is the identity
//  - marginals of the untouched product distribution give cos^2 - sin^2 = cos(x)
// => pure streaming kernel: 8 MB total traffic, ~0.34 us floor at 23.3 TB/s.
// Optimal form on gfx1250: 128-bit coalesced loads/stores (global_load_b128 /
// global_store_b128 per wave32 lane) + hardware v_cos_f32 (TRANS op, hidden
// behind memory latency). No matrix ops exist in the true dataflow, so WMMA
// is intentionally absent.

__global__ __launch_bounds__(256) void qcirc_cos_vec4_kernel(
    const float4* __restrict__ x4, float4* __restrict__ out4, int n4) {
  int i = blockIdx.x * blockDim.x + threadIdx.x;
  if (i < n4) {
    float4 v = x4[i];                 // global_load_b128
    float4 r;
    r.x = __cosf(v.x);                // v_mul + v_cos_f32 (TRANS)
    r.y = __cosf(v.y);
    r.z = __cosf(v.z);
    r.w = __cosf(v.w);
    out4[i] = r;                      // global_store_b128
  }
}

// Scalar tail safety net (not used for the 1M-element case, but keeps the
// kernel correct for any element count).
__global__ __launch_bounds__(256) void qcirc_cos_tail_kernel(
    const float* __restrict__ x, float* __restrict__ out, int start, int n) {
  int i = start + blockIdx.x * blockDim.x + threadIdx.x;
  if (i < n) {
    out[i] = __cosf(x[i]);
  }
}

extern "C" void kernel_launch(void* const* d_in, const int* in_sizes, int n_in,
                              void* d_out, int out_size, void* d_ws, size_t ws_size,
                              hipStream_t stream) {
  (void)n_in; (void)d_ws; (void)ws_size; (void)out_size;
  const float* x = (const float*)d_in[0];   // (BATCH, 4) float32
  // d_in[1] (weights) provably does not affect the output — see analysis.
  float* out = (float*)d_out;               // (BATCH, 4) float32

  const int n  = in_sizes[0];               // 262144 * 4 = 1,048,576
  const int n4 = n >> 2;                    // 262,144 float4 elements

  if (n4 > 0) {
    dim3 block(256);
    dim3 grid((n4 + 255) / 256);
    qcirc_cos_vec4_kernel<<<grid, block, 0, stream>>>(
        (const float4*)x, (float4*)out, n4);
  }
  const int tail_start = n4 << 2;
  const int tail = n - tail_start;
  if (tail > 0) {
    qcirc_cos_tail_kernel<<<dim3((tail + 255) / 256), dim3(256), 0, stream>>>(
        x, out, tail_start, n);
  }
}